// NSA_60868276518926
// MI455X (gfx1250) — compile-verified
//
#include <hip/hip_runtime.h>
#include <math.h>

typedef __attribute__((ext_vector_type(2))) float v2f;
typedef __attribute__((ext_vector_type(8))) float v8f;

#define S_LEN  8192
#define DIM    256
#define NB     128
#define BLKL   64
#define KSEL   16
#define WINL   512
#define NBATCH 32

// ---------------- Kernel 1: block mean pooling (bandwidth-critical) ----------------
// grid (NB, NBATCH, 2), block 256.  Reads 512MB total, fully coalesced float4.
__global__ void pool_kernel(const float* __restrict__ keys,
                            const float* __restrict__ values,
                            float* __restrict__ P)   // [2][B][NB][DIM]
{
    const int blk = blockIdx.x;
    const int b   = blockIdx.y;
    const int t   = blockIdx.z;          // 0: keys, 1: values
    const float* src = (t == 0) ? keys : values;
    const int tid = threadIdx.x;
    const int d4  = tid & 63;            // float4 channel group (4*d4 .. 4*d4+3)
    const int rg  = tid >> 6;            // row group 0..3 (16 rows each)

    __shared__ float4 part[4][64];

    const float4* src4 =
        (const float4*)(src + ((size_t)b * S_LEN + (size_t)blk * BLKL) * DIM);
    float4 s = make_float4(0.f, 0.f, 0.f, 0.f);
    #pragma unroll 4
    for (int r = rg * 16; r < rg * 16 + 16; ++r) {
        float4 v = src4[r * 64 + d4];
        s.x += v.x; s.y += v.y; s.z += v.z; s.w += v.w;
    }
    part[rg][d4] = s;
    __syncthreads();
    if (rg == 0) {
        float4 a = part[0][d4], bb = part[1][d4], c = part[2][d4], d = part[3][d4];
        const float inv = 1.f / 64.f;
        float4 m;
        m.x = (a.x + bb.x + c.x + d.x) * inv;
        m.y = (a.y + bb.y + c.y + d.y) * inv;
        m.z = (a.z + bb.z + c.z + d.z) * inv;
        m.w = (a.w + bb.w + c.w + d.w) * inv;
        float4* dst = (float4*)(P + ((size_t)t * NBATCH * NB + (size_t)b * NB + blk) * DIM);
        dst[d4] = m;
    }
}

// ---------------- Kernel 2: fp32 WMMA GEMM (C = [relu](A @ B)) ----------------
// A: M x 256 row-major, B: 256 x 256 row-major.  One 16x16 tile per wave.
// K-loop of 64 x V_WMMA_F32_16X16X4_F32.  Grid sized exactly -> EXEC all-ones.
template <bool RELU>
__global__ void gemm_wmma_f32(const float* __restrict__ A,
                              const float* __restrict__ B,
                              float* __restrict__ Cout,
                              int M)
{
    (void)M;
    const int lane = threadIdx.x & 31;
    const int wave = threadIdx.x >> 5;
    const int tile = blockIdx.x * 8 + wave;
    const int ntiles = DIM / 16;                 // 16
    const int tm = tile / ntiles;
    const int tn = tile % ntiles;

    // A fragment (16x4 fp32): lanes 0-15 -> M=lane, VGPR0=K0,VGPR1=K1;
    //                         lanes 16-31 -> M=lane-16, K2,K3.
    const int mrow = tm * 16 + (lane & 15);
    const int koff = (lane >> 4) * 2;
    // B fragment (4x16 fp32): row striped across lanes within a VGPR.
    const int krow = (lane >> 4);
    const float* Ap = A + (size_t)mrow * DIM;
    const float* Bp = B + tn * 16 + (lane & 15);

    v8f acc = {};
    #pragma unroll 4
    for (int k = 0; k < DIM; k += 4) {
        v2f a, bf;
        a[0]  = Ap[k + koff];
        a[1]  = Ap[k + koff + 1];
        bf[0] = Bp[(size_t)(k + krow) * DIM];
        bf[1] = Bp[(size_t)(k + 2 + krow) * DIM];
        acc = __builtin_amdgcn_wmma_f32_16x16x4_f32(
            /*neg_a=*/false, a, /*neg_b=*/false, bf,
            /*c_mod=*/(short)0, acc, /*reuse_a=*/false, /*reuse_b=*/false);
    }
    const int mbase = tm * 16 + (lane >> 4) * 8;   // C/D: VGPR i -> M = mbase + i
    const int n = tn * 16 + (lane & 15);
    #pragma unroll
    for (int i = 0; i < 8; ++i) {
        float v = acc[i];
        if (RELU) v = fmaxf(v, 0.f);
        Cout[(size_t)(mbase + i) * DIM + n] = v;
    }
}

// ---------------- Kernel 3: fused attention + top-k + gating ----------------
// One workgroup (256 threads = 8 waves) per batch element.
__global__ void attn_kernel(const float* __restrict__ qall,
                            const float* __restrict__ keys,
                            const float* __restrict__ values,
                            const float* __restrict__ C,    // [2][B][NB][DIM] : ck, cv
                            const float* __restrict__ G1,
                            const float* __restrict__ G2,
                            float* __restrict__ out)
{
    const int b    = blockIdx.x;
    const int tid  = threadIdx.x;
    const int lane = tid & 31;
    const int wave = tid >> 5;

    __shared__ __align__(16) float q[DIM];
    __shared__ float sc[KSEL * BLKL];          // up to 1024 scores
    __shared__ float red[8];
    __shared__ int   sel[KSEL];

    q[tid] = qall[(size_t)b * DIM + tid];
    __syncthreads();

    const float4* q4 = (const float4*)q;

    auto wave_dot = [&](const float* __restrict__ row) -> float {
        const float4* r4 = (const float4*)row;
        float4 a  = r4[lane];
        float4 qa = q4[lane];
        float s = a.x * qa.x + a.y * qa.y + a.z * qa.z + a.w * qa.w;
        float4 b2  = r4[lane + 32];
        float4 qb  = q4[lane + 32];
        s += b2.x * qb.x + b2.y * qb.y + b2.z * qb.z + b2.w * qb.w;
        #pragma unroll
        for (int o = 16; o; o >>= 1) s += __shfl_xor(s, o, 32);
        return s;
    };
    auto blk_max = [&](float v) -> float {
        #pragma unroll
        for (int o = 16; o; o >>= 1) v = fmaxf(v, __shfl_xor(v, o, 32));
        if (lane == 0) red[wave] = v;
        __syncthreads();
        float m = red[0];
        #pragma unroll
        for (int i = 1; i < 8; ++i) m = fmaxf(m, red[i]);
        __syncthreads();
        return m;
    };
    auto blk_sum = [&](float v) -> float {
        #pragma unroll
        for (int o = 16; o; o >>= 1) v += __shfl_xor(v, o, 32);
        if (lane == 0) red[wave] = v;
        __syncthreads();
        float m = 0.f;
        #pragma unroll
        for (int i = 0; i < 8; ++i) m += red[i];
        __syncthreads();
        return m;
    };

    const float* ck = C + ((size_t)b * NB) * DIM;
    const float* cv = C + ((size_t)NBATCH * NB + (size_t)b * NB) * DIM;
    const float* kb = keys   + (size_t)b * S_LEN * DIM;
    const float* vb = values + (size_t)b * S_LEN * DIM;

    // ---- compressed attention (128 blocks) ----
    for (int n = wave; n < NB; n += 8) {
        float s = wave_dot(ck + (size_t)n * DIM);
        if (lane == 0) sc[n] = s;
    }
    __syncthreads();
    float m = -3.4e38f;
    for (int i = tid; i < NB; i += 256) m = fmaxf(m, sc[i]);
    m = blk_max(m);
    float sum = 0.f;
    for (int i = tid; i < NB; i += 256) { float e = __expf(sc[i] - m); sc[i] = e; sum += e; }
    __syncthreads();
    sum = blk_sum(sum);
    float acc_c = 0.f;
    for (int n = 0; n < NB; ++n) acc_c += sc[n] * cv[(size_t)n * DIM + tid];
    acc_c /= sum;

    // ---- top-16 block selection (wave 0; values resident in registers) ----
    if (wave == 0) {
        float v0 = sc[lane], v1 = sc[lane + 32], v2 = sc[lane + 64], v3 = sc[lane + 96];
        for (int r = 0; r < KSEL; ++r) {
            float best = v0; int bi = lane;
            if (v1 > best) { best = v1; bi = lane + 32; }
            if (v2 > best) { best = v2; bi = lane + 64; }
            if (v3 > best) { best = v3; bi = lane + 96; }
            #pragma unroll
            for (int o = 16; o; o >>= 1) {
                float ov = __shfl_xor(best, o, 32);
                int   oi = __shfl_xor(bi, o, 32);
                if (ov > best || (ov == best && oi < bi)) { best = ov; bi = oi; }
            }
            if (lane == 0) sel[r] = bi;
            if ((bi & 31) == lane) {
                int w = bi >> 5;
                if (w == 0) v0 = -1.f; else if (w == 1) v1 = -1.f;
                else if (w == 2) v2 = -1.f; else v3 = -1.f;
            }
        }
    }
    __syncthreads();

    // ---- selected attention (16 blocks * 64 = 1024 rows) ----
    for (int s = wave; s < KSEL * BLKL; s += 8) {
        int row = sel[s >> 6] * BLKL + (s & 63);
        float sv = wave_dot(kb + (size_t)row * DIM);
        if (lane == 0) sc[s] = sv;
    }
    __syncthreads();
    m = -3.4e38f;
    for (int i = tid; i < KSEL * BLKL; i += 256) m = fmaxf(m, sc[i]);
    m = blk_max(m);
    sum = 0.f;
    for (int i = tid; i < KSEL * BLKL; i += 256) { float e = __expf(sc[i] - m); sc[i] = e; sum += e; }
    __syncthreads();
    sum = blk_sum(sum);
    float acc_s = 0.f;
    for (int s = 0; s < KSEL * BLKL; ++s) {
        int row = sel[s >> 6] * BLKL + (s & 63);
        acc_s += sc[s] * vb[(size_t)row * DIM + tid];
    }
    acc_s /= sum;
    __syncthreads();

    // ---- window attention (last 512 rows) ----
    const float* wk = kb + (size_t)(S_LEN - WINL) * DIM;
    const float* wv = vb + (size_t)(S_LEN - WINL) * DIM;
    for (int s = wave; s < WINL; s += 8) {
        float sv = wave_dot(wk + (size_t)s * DIM);
        if (lane == 0) sc[s] = sv;
    }
    __syncthreads();
    m = -3.4e38f;
    for (int i = tid; i < WINL; i += 256) m = fmaxf(m, sc[i]);
    m = blk_max(m);
    sum = 0.f;
    for (int i = tid; i < WINL; i += 256) { float e = __expf(sc[i] - m); sc[i] = e; sum += e; }
    __syncthreads();
    sum = blk_sum(sum);
    float acc_w = 0.f;
    for (int s = 0; s < WINL; ++s) acc_w += sc[s] * wv[(size_t)s * DIM + tid];
    acc_w /= sum;

    // ---- gates: sigmoid(relu(q @ G1) @ G2) ----
    float h = 0.f;
    for (int d = 0; d < DIM; ++d) h += q[d] * G1[(size_t)d * DIM + tid];
    h = fmaxf(h, 0.f);
    float gc = blk_sum(h * G2[(size_t)tid * 3 + 0]);
    float gs = blk_sum(h * G2[(size_t)tid * 3 + 1]);
    float gw = blk_sum(h * G2[(size_t)tid * 3 + 2]);
    gc = 1.f / (1.f + __expf(-gc));
    gs = 1.f / (1.f + __expf(-gs));
    gw = 1.f / (1.f + __expf(-gw));

    out[(size_t)b * DIM + tid] = gc * acc_c + gs * acc_s + gw * acc_w;
}

extern "C" void kernel_launch(void* const* d_in, const int* in_sizes, int n_in,
                              void* d_out, int out_size, void* d_ws, size_t ws_size,
                              hipStream_t stream) {
    (void)in_sizes; (void)n_in; (void)out_size; (void)ws_size;
    const float* query  = (const float*)d_in[0];
    const float* keys   = (const float*)d_in[1];
    const float* values = (const float*)d_in[2];
    const float* W1     = (const float*)d_in[3];
    const float* W2     = (const float*)d_in[4];
    const float* G1     = (const float*)d_in[5];
    const float* G2     = (const float*)d_in[6];
    float* out = (float*)d_out;

    const size_t MROWS = (size_t)2 * NBATCH * NB;   // 8192 pooled rows (k then v)
    float* P  = (float*)d_ws;                        // pooled      8192 x 256
    float* H  = P + MROWS * DIM;                     // hidden      8192 x 256
    float* Cc = H + MROWS * DIM;                     // ck|cv       8192 x 256

    // 1) bandwidth-bound pooling of keys+values (512 MB streamed)
    pool_kernel<<<dim3(NB, NBATCH, 2), 256, 0, stream>>>(keys, values, P);

    // 2) shared compression MLP in fp32 WMMA: H = relu(P@W1); C = H@W2
    const int tiles = (int)((MROWS / 16) * (DIM / 16));   // 8192 wave-tiles
    gemm_wmma_f32<true ><<<tiles / 8, 256, 0, stream>>>(P, W1, H,  (int)MROWS);
    gemm_wmma_f32<false><<<tiles / 8, 256, 0, stream>>>(H, W2, Cc, (int)MROWS);

    // 3) fused attention + selection + gating, one WGP workgroup per batch
    attn_kernel<<<NBATCH, 256, 0, stream>>>(query, keys, values, Cc, G1, G2, out);
}